// WordSAGE_52123723104477
// MI455X (gfx1250) — compile-verified
//
#include <hip/hip_runtime.h>
#include <hip/hip_bf16.h>

// ---------------------------------------------------------------------------
// WordSAGE on MI455X (gfx1250): 2-layer mean-aggregation GraphSAGE + classifier
//   h1 = relu(x @ Ws1 + mean_agg(x) @ Wn1 + b1)
//   h2 = relu(h1 @ Ws2 + mean_agg(h1) @ Wn2 + b2)
//   out = h2 @ Wcls + bcls                      [N x 40] fp32
// GEMMs use V_WMMA_F32_16X16X4_F32 (fp32 in/out -> matches fp32 reference).
// GEMM kernels run one wave32 per workgroup so the row-tile base (m0) is an
// SGPR: all bounds checks lower to scalar s_cmp/s_cbranch and the full-stripe
// store path is branch-free with EXEC all-1s (WMMA requirement).
// Each GEMM wave computes a 32-row x 128-col stripe (16 v8f accumulators) so
// every weight-fragment load feeds two WMMAs. Aggregation uses wave-per-edge
// float4 gather + global_atomic_add_f32 (51 MB feature table is L2-resident
// in the 192 MB L2, so the 3.2 GB of edge traffic stays on-chip).
// ---------------------------------------------------------------------------

typedef float v2f __attribute__((ext_vector_type(2)));
typedef float v8f __attribute__((ext_vector_type(8)));

__device__ __forceinline__ v8f wmma_f32x4(v2f a, v2f b, v8f c) {
    // 8 args: (neg_a, A, neg_b, B, c_mod, C, reuse_a, reuse_b)
    return __builtin_amdgcn_wmma_f32_16x16x4_f32(
        false, a, false, b, (short)0, c, false, false);
}

#define C128 128

// ---------------- degree: deg[dst] += 1 per edge ---------------------------
__global__ __launch_bounds__(256) void deg_kernel(const int* __restrict__ dst,
                                                  float* __restrict__ deg,
                                                  int nedges) {
    int e = blockIdx.x * 256 + threadIdx.x;
    if (e < nedges) {
        unsafeAtomicAdd(&deg[dst[e]], 1.0f);
    }
}

// ---------------- aggregation: sums[dst,:] += h[src,:] ---------------------
// One wave32 per edge; lane l moves channels [4l, 4l+3] (float4 = full row).
__global__ __launch_bounds__(256) void agg_kernel(const float* __restrict__ H,
                                                  const int* __restrict__ src,
                                                  const int* __restrict__ dst,
                                                  float* __restrict__ Sums,
                                                  int nedges) {
    int e = blockIdx.x * 8 + (threadIdx.x >> 5);
    if (e >= nedges) return;
    int lane = threadIdx.x & 31;
    int s = src[e];
    int d = dst[e];
    const float4* hp = reinterpret_cast<const float4*>(H + (size_t)s * C128);
    float4 v = hp[lane];
    float* op = Sums + (size_t)d * C128 + lane * 4;
    unsafeAtomicAdd(op + 0, v.x);
    unsafeAtomicAdd(op + 1, v.y);
    unsafeAtomicAdd(op + 2, v.z);
    unsafeAtomicAdd(op + 3, v.w);
}

// ---------------- fused SAGE layer GEMM ------------------------------------
// out[i,:] = relu( H[i,:] @ Ws  +  (Sums[i,:]/max(deg[i],1)) @ Wn  + bias )
// One wave per workgroup; wave handles a 32-row x 128-col stripe:
// two M-tiles x 8 N-tiles of 16x16, K-loop in steps of 4; each B fragment is
// reused by both M-tiles.
__global__ __launch_bounds__(32) void sage_gemm_kernel(
    const float* __restrict__ H, const float* __restrict__ Sums,
    const float* __restrict__ Deg, const float* __restrict__ Ws,
    const float* __restrict__ Wn, const float* __restrict__ bias,
    float* __restrict__ Out, int nrows) {
    const int lane = threadIdx.x;        // 0..31
    const int half = lane >> 4;          // 0: lanes 0-15, 1: lanes 16-31
    const int lr   = lane & 15;
    const int m0   = blockIdx.x * 32;    // SGPR -> scalar bounds checks
    if (m0 >= nrows) return;

    // A-matrix rows held by this lane for the two M-tiles (same for both halves)
    int row0 = m0 + lr;
    int row1 = m0 + 16 + lr;
    if (row0 >= nrows) row0 = nrows - 1;
    if (row1 >= nrows) row1 = nrows - 1;
    const float invd0 = 1.0f / fmaxf(Deg[row0], 1.0f);
    const float invd1 = 1.0f / fmaxf(Deg[row1], 1.0f);

    v8f acc0[8], acc1[8];
#pragma unroll
    for (int nt = 0; nt < 8; ++nt) {
        float bv = bias[nt * 16 + lr];
#pragma unroll
        for (int i = 0; i < 8; ++i) { acc0[nt][i] = bv; acc1[nt][i] = bv; }
    }

    const float* arow0 = H    + (size_t)row0 * C128;
    const float* nrow0 = Sums + (size_t)row0 * C128;
    const float* arow1 = H    + (size_t)row1 * C128;
    const float* nrow1 = Sums + (size_t)row1 * C128;

    for (int k0 = 0; k0 < C128; k0 += 4) {
        const int kk = k0 + 2 * half;
        v2f as0, an0, as1, an1;
        as0.x = arow0[kk];         as0.y = arow0[kk + 1];
        an0.x = nrow0[kk] * invd0; an0.y = nrow0[kk + 1] * invd0;
        as1.x = arow1[kk];         as1.y = arow1[kk + 1];
        an1.x = nrow1[kk] * invd1; an1.y = nrow1[kk + 1] * invd1;
#pragma unroll
        for (int nt = 0; nt < 8; ++nt) {
            const int col = nt * 16 + lr;
            v2f bs, bn;
            bs.x = Ws[(size_t)kk * C128 + col];
            bs.y = Ws[(size_t)(kk + 1) * C128 + col];
            bn.x = Wn[(size_t)kk * C128 + col];
            bn.y = Wn[(size_t)(kk + 1) * C128 + col];
            acc0[nt] = wmma_f32x4(as0, bs, acc0[nt]);
            acc1[nt] = wmma_f32x4(as1, bs, acc1[nt]);
            acc0[nt] = wmma_f32x4(an0, bn, acc0[nt]);
            acc1[nt] = wmma_f32x4(an1, bn, acc1[nt]);
        }
    }

    // D layout: VGPR r holds M = r + 8*half, N = lr (within a 16x16 tile)
    if (m0 + 32 <= nrows) {          // scalar branch: full stripe, no masking
#pragma unroll
        for (int nt = 0; nt < 8; ++nt) {
#pragma unroll
            for (int r = 0; r < 8; ++r) {
                int ra = m0 + r + 8 * half;
                int rb = ra + 16;
                Out[(size_t)ra * C128 + nt * 16 + lr] = fmaxf(acc0[nt][r], 0.0f);
                Out[(size_t)rb * C128 + nt * 16 + lr] = fmaxf(acc1[nt][r], 0.0f);
            }
        }
    } else {                          // tail stripe (last block only)
#pragma unroll
        for (int nt = 0; nt < 8; ++nt) {
#pragma unroll
            for (int r = 0; r < 8; ++r) {
                int ra = m0 + r + 8 * half;
                int rb = ra + 16;
                if (ra < nrows)
                    Out[(size_t)ra * C128 + nt * 16 + lr] = fmaxf(acc0[nt][r], 0.0f);
                if (rb < nrows)
                    Out[(size_t)rb * C128 + nt * 16 + lr] = fmaxf(acc1[nt][r], 0.0f);
            }
        }
    }
}

// ---------------- classifier: out = H @ Wc + bc   (N_CLS = 40) -------------
__global__ __launch_bounds__(32) void cls_gemm_kernel(
    const float* __restrict__ H, const float* __restrict__ Wc,
    const float* __restrict__ bc, float* __restrict__ Out, int nrows) {
    const int NCLS = 40;
    const int lane = threadIdx.x;
    const int half = lane >> 4;
    const int lr   = lane & 15;
    const int m0   = blockIdx.x * 16;    // SGPR
    if (m0 >= nrows) return;

    int rowA = m0 + lr;
    if (rowA >= nrows) rowA = nrows - 1;

    v8f acc[3];
    int colc[3];
#pragma unroll
    for (int nt = 0; nt < 3; ++nt) {
        int col = nt * 16 + lr;
        colc[nt] = col < NCLS ? col : NCLS - 1;  // clamp loads in-bounds
        float bv = bc[colc[nt]];
#pragma unroll
        for (int i = 0; i < 8; ++i) acc[nt][i] = bv;
    }

    const float* arow = H + (size_t)rowA * C128;
    for (int k0 = 0; k0 < C128; k0 += 4) {
        const int kk = k0 + 2 * half;
        v2f a;
        a.x = arow[kk];
        a.y = arow[kk + 1];
#pragma unroll
        for (int nt = 0; nt < 3; ++nt) {
            v2f b;
            b.x = Wc[(size_t)kk * NCLS + colc[nt]];
            b.y = Wc[(size_t)(kk + 1) * NCLS + colc[nt]];
            acc[nt] = wmma_f32x4(a, b, acc[nt]);
        }
    }

    if (m0 + 16 <= nrows) {           // scalar branch: full row-tile
#pragma unroll
        for (int nt = 0; nt < 3; ++nt) {
            int col = nt * 16 + lr;
            if (col < NCLS) {         // lane predicate only on last N-tile
#pragma unroll
                for (int r = 0; r < 8; ++r) {
                    int row = m0 + r + 8 * half;
                    Out[(size_t)row * NCLS + col] = acc[nt][r];
                }
            }
        }
    } else {                          // tail rows (last block only)
#pragma unroll
        for (int nt = 0; nt < 3; ++nt) {
            int col = nt * 16 + lr;
            if (col < NCLS) {
#pragma unroll
                for (int r = 0; r < 8; ++r) {
                    int row = m0 + r + 8 * half;
                    if (row < nrows)
                        Out[(size_t)row * NCLS + col] = acc[nt][r];
                }
            }
        }
    }
}

// ---------------------------------------------------------------------------
extern "C" void kernel_launch(void* const* d_in, const int* in_sizes, int n_in,
                              void* d_out, int out_size, void* d_ws, size_t ws_size,
                              hipStream_t stream) {
    const float* x      = (const float*)d_in[0];
    const int*   ei     = (const int*)d_in[1];   // [2, E] row-major: src then dst
    const float* Wself1 = (const float*)d_in[2];
    const float* Wneigh1= (const float*)d_in[3];
    const float* b1     = (const float*)d_in[4];
    const float* Wself2 = (const float*)d_in[5];
    const float* Wneigh2= (const float*)d_in[6];
    const float* b2     = (const float*)d_in[7];
    const float* Wcls   = (const float*)d_in[8];
    const float* bcls   = (const float*)d_in[9];

    const int N = in_sizes[0] / C128;     // 100000 nodes
    const int E = in_sizes[1] / 2;        // 1600000 edges
    const int* src = ei;
    const int* dst = ei + E;

    // workspace layout (512B-aligned regions)
    char* ws = (char*)d_ws;
    size_t degB  = ((size_t)N * sizeof(float) + 511) & ~(size_t)511;
    size_t featB = (((size_t)N * C128 * sizeof(float)) + 511) & ~(size_t)511;
    float* deg  = (float*)(ws);
    float* sums = (float*)(ws + degB);
    float* h1   = (float*)(ws + degB + featB);
    float* h2   = (float*)(ws + degB + 2 * featB);

    const int edgeThreadBlocks = (E + 255) / 256;  // deg: 1 thread/edge
    const int edgeWaveBlocks   = (E + 7) / 8;      // agg: 1 wave/edge, 8 waves/block
    const int sageBlocks       = (N + 31) / 32;    // sage gemm: 1 wave = 32 rows
    const int clsBlocks        = (N + 15) / 16;    // cls gemm: 1 wave = 16 rows

    // --- degrees (shared by both layers) + layer-1 aggregation -------------
    hipMemsetAsync(deg, 0, (size_t)N * sizeof(float), stream);
    hipMemsetAsync(sums, 0, (size_t)N * C128 * sizeof(float), stream);
    deg_kernel<<<edgeThreadBlocks, 256, 0, stream>>>(dst, deg, E);
    agg_kernel<<<edgeWaveBlocks, 256, 0, stream>>>(x, src, dst, sums, E);
    sage_gemm_kernel<<<sageBlocks, 32, 0, stream>>>(x, sums, deg, Wself1,
                                                    Wneigh1, b1, h1, N);

    // --- layer 2 ------------------------------------------------------------
    hipMemsetAsync(sums, 0, (size_t)N * C128 * sizeof(float), stream);
    agg_kernel<<<edgeWaveBlocks, 256, 0, stream>>>(h1, src, dst, sums, E);
    sage_gemm_kernel<<<sageBlocks, 32, 0, stream>>>(h1, sums, deg, Wself2,
                                                    Wneigh2, b2, h2, N);

    // --- classifier ---------------------------------------------------------
    cls_gemm_kernel<<<clsBlocks, 32, 0, stream>>>(h2, Wcls, bcls,
                                                  (float*)d_out, N);
}